// AdaptiveAxisAttention_39204461478398
// MI455X (gfx1250) — compile-verified
//
#include <hip/hip_runtime.h>
#include <hip/hip_bf16.h>
#include <math.h>

#define EPSF 1e-5f

typedef float v2f __attribute__((ext_vector_type(2)));
typedef float v8f __attribute__((ext_vector_type(8)));

// ---------------------------------------------------------------------------
// CDNA5 WMMA f32 16x16x4:  D(16x16,f32) = A(16x4,f32) * B(4x16,f32) + C
// A: lane m=lane&15 holds row M=m; VGPR0 = K=2*kh, VGPR1 = K=2*kh+1 (kh=lane>>4)
// B: VGPR0 = row K=2*kh cols 0..15 (lane&15), VGPR1 = row K=2*kh+1
// C/D: VGPR r holds D[M = r + 8*kh][N = lane&15]
// ---------------------------------------------------------------------------
__device__ __forceinline__ v8f wmma4(v2f a, v2f b, v8f c) {
  return __builtin_amdgcn_wmma_f32_16x16x4_f32(false, a, false, b, (short)0, c,
                                               false, false);
}

// half-pixel linear interpolation coordinates
__device__ __forceinline__ void lin_coord(int i, int n_out, int n_in, int& i0,
                                          int& i1, float& f) {
  float s = (i + 0.5f) * ((float)n_in / (float)n_out) - 0.5f;
  s = fmaxf(s, 0.0f);
  int a = (int)s;
  if (a > n_in - 1) a = n_in - 1;
  i0 = a;
  i1 = (a + 1 < n_in) ? a + 1 : n_in - 1;
  f = fminf(fmaxf(s - (float)a, 0.0f), 1.0f);
}

__device__ __forceinline__ void block_reduce2(float& s, float& q, float* ss,
                                              float* sq) {
  int tid = threadIdx.x;
  ss[tid] = s; sq[tid] = q;
  __syncthreads();
  for (int o = 128; o > 0; o >>= 1) {
    if (tid < o) { ss[tid] += ss[tid + o]; sq[tid] += sq[tid + o]; }
    __syncthreads();
  }
  s = ss[0]; q = sq[0];
}

// ===========================================================================
// 1) qkv[b,o,l] = sum_c qkv_w[o,c] * x[n,c,l,w]   (b = n*64+w)
//    WMMA GEMM, 2x2 register-blocked: 32x32 output tile per wave,
//    columns = 32 consecutive w (coalesced x loads)
// ===========================================================================
__global__ void k_qkv_gemm(const float* __restrict__ Wq,
                           const float* __restrict__ x,
                           float* __restrict__ qkv) {
  int wave = (blockIdx.x * 256 + threadIdx.x) >> 5;
  int lane = threadIdx.x & 31;
  int mT = wave & 7;           // 8 tiles of 32 rows over 256
  int colT = wave >> 3;        // 1024 column tiles of 32
  int m = lane & 15, kh = lane >> 4, nl = lane & 15;
  int m0 = mT * 32;
  int col0 = colT * 32;        // col = (n*64 + l)*64 + w, w fastest
  int w0 = col0 & 63;
  int l = (col0 >> 6) & 63;
  int n = col0 >> 12;
  size_t bb0 = (size_t)n * 524288 + (size_t)l * 64 + (w0 + nl);
  size_t bb1 = bb0 + 16;
  const float* ar0 = Wq + (size_t)(m0 + m) * 128;
  const float* ar1 = ar0 + 16 * 128;
  v8f a00 = {}, a01 = {}, a10 = {}, a11 = {};
  for (int k0 = 0; k0 < 128; k0 += 4) {
    int c = k0 + 2 * kh;
    v2f A0; A0.x = ar0[c]; A0.y = ar0[c + 1];
    v2f A1; A1.x = ar1[c]; A1.y = ar1[c + 1];
    v2f B0; B0.x = x[bb0 + (size_t)c * 4096];
    B0.y = x[bb0 + (size_t)(c + 1) * 4096];
    v2f B1; B1.x = x[bb1 + (size_t)c * 4096];
    B1.y = x[bb1 + (size_t)(c + 1) * 4096];
    a00 = wmma4(A0, B0, a00);
    a01 = wmma4(A0, B1, a01);
    a10 = wmma4(A1, B0, a10);
    a11 = wmma4(A1, B1, a11);
  }
  size_t b_idx0 = (size_t)n * 64 + w0 + nl;
  size_t ob0 = b_idx0 * 16384 + l;            // qkv[(b*256+oc)*64 + l]
  size_t ob1 = ob0 + 16 * 16384;
#pragma unroll
  for (int r = 0; r < 8; ++r) {
    int oc0 = m0 + r + 8 * kh;
    int oc1 = oc0 + 16;
    qkv[ob0 + (size_t)oc0 * 64] = a00[r];
    qkv[ob1 + (size_t)oc0 * 64] = a01[r];
    qkv[ob0 + (size_t)oc1 * 64] = a10[r];
    qkv[ob1 + (size_t)oc1 * 64] = a11[r];
  }
}

// ===========================================================================
// 2) per-channel mean/var over (B,L) for layout (B, CH, L)
// ===========================================================================
__global__ void k_chan_stats(const float* __restrict__ x,
                             float* __restrict__ stat, int CH, int Bn, int L) {
  __shared__ float ss[256], sq[256];
  int ch = blockIdx.x;
  int tot = Bn * L;
  float s = 0.f, q = 0.f;
  for (int idx = threadIdx.x; idx < tot; idx += 256) {
    int b = idx / L, l = idx - b * L;
    float v = x[((size_t)b * CH + ch) * L + l];
    s += v; q += v * v;
  }
  block_reduce2(s, q, ss, sq);
  if (threadIdx.x == 0) {
    float mu = s / (float)tot;
    stat[2 * ch] = mu;
    stat[2 * ch + 1] = q / (float)tot - mu * mu;
  }
}

// ===========================================================================
// 3) pos(32,33,33) = bilinear(base_relative(32,127,127))
// ===========================================================================
__global__ void k_pos(const float* __restrict__ br, float* __restrict__ pos) {
  int t = blockIdx.x * 256 + threadIdx.x;
  if (t >= 32 * 1089) return;
  int c = t / 1089, ij = t - c * 1089;
  int i = ij / 33, j = ij - i * 33;
  int y0, y1, x0, x1; float fy, fx;
  lin_coord(i, 33, 127, y0, y1, fy);
  lin_coord(j, 33, 127, x0, x1, fx);
  const float* s = br + (size_t)c * 127 * 127;
  float a = s[y0 * 127 + x0] + (s[y0 * 127 + x1] - s[y0 * 127 + x0]) * fx;
  float b = s[y1 * 127 + x0] + (s[y1 * 127 + x1] - s[y1 * 127 + x0]) * fx;
  pos[t] = a + (b - a) * fy;
}

// ===========================================================================
// 4) BN(qkv) then split q/k/v and linear-resize l 64->33
// ===========================================================================
__global__ void k_split_resize(const float* __restrict__ qkv,
                               const float* __restrict__ stat,
                               const float* __restrict__ gam,
                               const float* __restrict__ bet,
                               float* __restrict__ qa, float* __restrict__ ka,
                               float* __restrict__ va) {
  int t = blockIdx.x * 256 + threadIdx.x;   // 4325376 exact
  int i = t % 33;
  int r = t / 33;
  int cc = r & 31;
  int bg = r >> 5;          // b*8+g
  int g = bg & 7, b = bg >> 3;
  int ch = g * 32 + cc;
  int l0, l1; float f;
  lin_coord(i, 33, 64, l0, l1, f);
  const float* src = qkv + ((size_t)b * 256 + ch) * 64;
  float v = src[l0] + (src[l1] - src[l0]) * f;
  v = (v - stat[2 * ch]) * rsqrtf(stat[2 * ch + 1] + EPSF) * gam[ch] + bet[ch];
  if (cc < 8)
    qa[((size_t)bg * 8 + cc) * 33 + i] = v;
  else if (cc < 16)
    ka[((size_t)bg * 8 + (cc - 8)) * 33 + i] = v;
  else
    va[((size_t)bg * 16 + (cc - 16)) * 33 + i] = v;
}

// ===========================================================================
// 5) sim_pre(512,24,33,33): cc<8 qk | cc<16 qr | cc<24 kr (pre-transposed)
// ===========================================================================
__global__ void k_sim_pre(const float* __restrict__ qa,
                          const float* __restrict__ ka,
                          const float* __restrict__ pos,
                          float* __restrict__ simp) {
  int t = blockIdx.x * 256 + threadIdx.x;   // 13381632 exact
  int ij = t % 1089;
  int r = t / 1089;
  int cc = r % 24;
  int b = r / 24;
  int i = ij / 33, j = ij - i * 33;
  int g = cc & 7;
  const float* qab = qa + ((size_t)b * 8 + g) * 264;  // 8*33
  const float* kab = ka + ((size_t)b * 8 + g) * 264;
  float s = 0.f;
  if (cc < 8) {
#pragma unroll
    for (int c = 0; c < 8; ++c) s += qab[c * 33 + i] * kab[c * 33 + j];
  } else if (cc < 16) {
#pragma unroll
    for (int c = 0; c < 8; ++c) s += qab[c * 33 + i] * pos[(size_t)c * 1089 + ij];
  } else {
#pragma unroll
    for (int c = 0; c < 8; ++c)
      s += kab[c * 33 + j] * pos[(size_t)(8 + c) * 1089 + j * 33 + i];
  }
  simp[t] = s;
}

// ===========================================================================
// 6) sim stats per (n, cc) over (w,i,j)
// ===========================================================================
__global__ void k_sim_stats(const float* __restrict__ simp,
                            float* __restrict__ sstat) {
  __shared__ float ss[256], sq[256];
  int n = blockIdx.x / 24, cc = blockIdx.x - n * 24;
  const int tot = 64 * 1089;
  float s = 0.f, q = 0.f;
  for (int idx = threadIdx.x; idx < tot; idx += 256) {
    int w = idx / 1089, ij = idx - w * 1089;
    float v = simp[((size_t)(n * 64 + w) * 24 + cc) * 1089 + ij];
    s += v; q += v * v;
  }
  block_reduce2(s, q, ss, sq);
  if (threadIdx.x == 0) {
    float mu = s / (float)tot;
    sstat[2 * blockIdx.x] = mu;
    sstat[2 * blockIdx.x + 1] = q / (float)tot - mu * mu;
  }
}

// ===========================================================================
// 7) normalize sim + sum 3 groups -> simG(512,8,33,33)
// ===========================================================================
__global__ void k_simG(const float* __restrict__ simp,
                       const float* __restrict__ sstat,
                       const float* __restrict__ gam,
                       const float* __restrict__ bet,
                       float* __restrict__ simG) {
  int t = blockIdx.x * 256 + threadIdx.x;   // 4460544 exact
  int ij = t % 1089;
  int bg = t / 1089;
  int g = bg & 7, b = bg >> 3, n = b >> 6;
  float s = 0.f;
#pragma unroll
  for (int tt = 0; tt < 3; ++tt) {
    int ch = tt * 8 + g;
    float v = simp[((size_t)b * 24 + ch) * 1089 + ij];
    int si = n * 24 + ch;
    s += (v - sstat[2 * si]) * rsqrtf(sstat[2 * si + 1] + EPSF) * gam[ch] +
         bet[ch];
  }
  simG[t] = s;
}

// ===========================================================================
// 8) bilinear 33->64 + softmax(last dim) -> simH(512,8,64,64); 1 thread/row
// ===========================================================================
__global__ void k_resize_softmax(const float* __restrict__ simG,
                                 float* __restrict__ simH) {
  int t = blockIdx.x * 256 + threadIdx.x;   // 262144 exact: (b*8+g)*64 + i
  int i = t & 63;
  int bg = t >> 6;
  const float* src = simG + (size_t)bg * 1089;
  int y0, y1; float fy;
  lin_coord(i, 64, 33, y0, y1, fy);
  const float* r0 = src + y0 * 33;
  const float* r1 = src + y1 * 33;
  float mx = -1e30f;
  for (int j = 0; j < 64; ++j) {
    int x0, x1; float fx;
    lin_coord(j, 64, 33, x0, x1, fx);
    float a = r0[x0] + (r0[x1] - r0[x0]) * fx;
    float b = r1[x0] + (r1[x1] - r1[x0]) * fx;
    mx = fmaxf(mx, a + (b - a) * fy);
  }
  float sum = 0.f;
  for (int j = 0; j < 64; ++j) {
    int x0, x1; float fx;
    lin_coord(j, 64, 33, x0, x1, fx);
    float a = r0[x0] + (r0[x1] - r0[x0]) * fx;
    float b = r1[x0] + (r1[x1] - r1[x0]) * fx;
    sum += __expf((a + (b - a) * fy) - mx);
  }
  float inv = 1.f / sum;
  float* dst = simH + (size_t)t * 64;
  for (int j = 0; j < 64; ++j) {
    int x0, x1; float fx;
    lin_coord(j, 64, 33, x0, x1, fx);
    float a = r0[x0] + (r0[x1] - r0[x0]) * fx;
    float b = r1[x0] + (r1[x1] - r1[x0]) * fx;
    dst[j] = __expf((a + (b - a) * fy) - mx) * inv;
  }
}

// ===========================================================================
// 9) vb(512,8,16,64) = linear resize of va last dim 33->64
// ===========================================================================
__global__ void k_vb(const float* __restrict__ va, float* __restrict__ vb) {
  int t = blockIdx.x * 256 + threadIdx.x;   // 4194304 exact
  int l = t & 63;
  int bgc = t >> 6;                         // (b*8+g)*16 + c
  int l0, l1; float f;
  lin_coord(l, 64, 33, l0, l1, f);
  const float* s = va + (size_t)bgc * 33;
  vb[t] = s[l0] + (s[l1] - s[l0]) * f;
}

// ===========================================================================
// 10) v_eH(16,64,64) = bilinear(pos[16..31], 33->64)
// ===========================================================================
__global__ void k_veH(const float* __restrict__ pos, float* __restrict__ veH) {
  int t = blockIdx.x * 256 + threadIdx.x;   // 65536 exact
  int c = t >> 12;
  int pix = t & 4095;
  int i = pix >> 6, j = pix & 63;
  int y0, y1, x0, x1; float fy, fx;
  lin_coord(i, 64, 33, y0, y1, fy);
  lin_coord(j, 64, 33, x0, x1, fx);
  const float* s = pos + (size_t)(16 + c) * 1089;
  float a = s[y0 * 33 + x0] + (s[y0 * 33 + x1] - s[y0 * 33 + x0]) * fx;
  float b = s[y1 * 33 + x0] + (s[y1 * 33 + x1] - s[y1 * 33 + x0]) * fx;
  veH[t] = a + (b - a) * fy;
}

// ===========================================================================
// 11) sv: per (b,g)  D[i,c] = sum_j sim[i,j]*vb[c,j]  -> so even channels
//     WMMA: M=i (2x1 blocked: 32 rows/wave), N=c(16), K=j(64)
// ===========================================================================
__global__ void k_sv(const float* __restrict__ simH,
                     const float* __restrict__ vb, float* __restrict__ so) {
  int wave = (blockIdx.x * 256 + threadIdx.x) >> 5;
  int lane = threadIdx.x & 31;
  int iT = wave & 1;                        // 2 tiles of 32 rows
  int bg = wave >> 1;                       // b*8+g, 4096 total
  int m = lane & 15, kh = lane >> 4, nl = lane & 15;
  int i0 = iT * 32;
  const float* sb = simH + (size_t)bg * 4096;
  const float* vbb = vb + (size_t)bg * 1024;
  v8f a0 = {}, a1 = {};
  for (int k0 = 0; k0 < 64; k0 += 4) {
    int j = k0 + 2 * kh;
    v2f A0; A0.x = sb[(i0 + m) * 64 + j]; A0.y = sb[(i0 + m) * 64 + j + 1];
    v2f A1; A1.x = sb[(i0 + 16 + m) * 64 + j];
    A1.y = sb[(i0 + 16 + m) * 64 + j + 1];
    v2f B; B.x = vbb[nl * 64 + j]; B.y = vbb[nl * 64 + j + 1];
    a0 = wmma4(A0, B, a0);
    a1 = wmma4(A1, B, a1);
  }
  int b = bg >> 3, g = bg & 7;
  size_t obase = (size_t)b * 16384 + (size_t)(2 * (g * 16 + nl)) * 64;
#pragma unroll
  for (int r = 0; r < 8; ++r) {
    int i = i0 + r + 8 * kh;
    so[obase + i] = a0[r];
    so[obase + i + 16] = a1[r];
  }
}

// ===========================================================================
// 12) sve: out[c,i] = sum_j sim[i,j]*veH[c,i,j] -> so odd channels
// ===========================================================================
__global__ void k_sve(const float* __restrict__ simH,
                      const float* __restrict__ veH, float* __restrict__ so) {
  int t = blockIdx.x * 256 + threadIdx.x;   // 4194304 exact
  int i = t & 63;
  int r = t >> 6;
  int c = r & 15;
  int bg = r >> 4;
  const float* sr = simH + ((size_t)bg * 64 + i) * 64;
  const float* vr = veH + ((size_t)c * 64 + i) * 64;
  float s = 0.f;
  for (int j = 0; j < 64; ++j) s += sr[j] * vr[j];
  int b = bg >> 3, g = bg & 7;
  so[((size_t)b * 256 + 2 * (g * 16 + c) + 1) * 64 + i] = s;
}

// ===========================================================================
// 13) BN(so) + pair-sum -> o(N,C,H,W)
// ===========================================================================
__global__ void k_o(const float* __restrict__ so,
                    const float* __restrict__ stat,
                    const float* __restrict__ gam,
                    const float* __restrict__ bet, float* __restrict__ o) {
  int t = blockIdx.x * 256 + threadIdx.x;   // 4194304: ((n*128+c)*64+l)*64+w
  int n = t >> 19;
  int c = (t >> 12) & 127;
  int pix = t & 4095;
  int l = pix >> 6, w = pix & 63;
  int b = n * 64 + w;
  int c0 = 2 * c, c1 = 2 * c + 1;
  float v0 = so[((size_t)b * 256 + c0) * 64 + l];
  float v1 = so[((size_t)b * 256 + c1) * 64 + l];
  v0 = (v0 - stat[2 * c0]) * rsqrtf(stat[2 * c0 + 1] + EPSF) * gam[c0] + bet[c0];
  v1 = (v1 - stat[2 * c1]) * rsqrtf(stat[2 * c1 + 1] + EPSF) * gam[c1] + bet[c1];
  o[t] = v0 + v1;
}

// ===========================================================================
// spatial shift helper (xo channels from o)
// ===========================================================================
__device__ __forceinline__ float xo_val(const float* __restrict__ o, int n,
                                        int c, int h, int w) {
  size_t nb = (size_t)n * 524288;
  if (c < 10) { int ws = w - 2; return (ws >= 0) ? o[nb + ((size_t)c * 64 + h) * 64 + ws] : 0.f; }
  if (c < 20) { int ws = w + 2; return (ws < 64) ? o[nb + ((size_t)(c - 10) * 64 + h) * 64 + ws] : 0.f; }
  if (c < 30) { int hs = h - 2; return (hs >= 0) ? o[nb + ((size_t)(c - 20) * 64 + hs) * 64 + w] : 0.f; }
  if (c < 40) { int hs = h + 2; return (hs < 64) ? o[nb + ((size_t)(c - 30) * 64 + hs) * 64 + w] : 0.f; }
  return o[nb + ((size_t)c * 64 + h) * 64 + w];
}

// 14) instance-norm stats per (n,c) over HW (on shifted xo)
__global__ void k_in_stats(const float* __restrict__ o,
                           float* __restrict__ stat) {
  __shared__ float ss[256], sq[256];
  int n = blockIdx.x >> 7, c = blockIdx.x & 127;
  float s = 0.f, q = 0.f;
  for (int p = threadIdx.x; p < 4096; p += 256) {
    float v = xo_val(o, n, c, p >> 6, p & 63);
    s += v; q += v * v;
  }
  block_reduce2(s, q, ss, sq);
  if (threadIdx.x == 0) {
    float mu = s / 4096.f;
    stat[2 * blockIdx.x] = mu;
    stat[2 * blockIdx.x + 1] = q / 4096.f - mu * mu;
  }
}

// 15) xn = instance-norm(xo)
__global__ void k_xn(const float* __restrict__ o, const float* __restrict__ stat,
                     const float* __restrict__ gam, const float* __restrict__ bet,
                     float* __restrict__ xn) {
  int t = blockIdx.x * 256 + threadIdx.x;   // 4194304 exact
  int n = t >> 19;
  int c = (t >> 12) & 127;
  int pix = t & 4095;
  float v = xo_val(o, n, c, pix >> 6, pix & 63);
  int si = n * 128 + c;
  xn[t] = (v - stat[2 * si]) * rsqrtf(stat[2 * si + 1] + EPSF) * gam[c] + bet[c];
}

// ===========================================================================
// 16) h1 = gelu(mlp_w1 @ xn)  WMMA GEMM 2x2-blocked, M=512, K=128
// ===========================================================================
__global__ void k_mlp1(const float* __restrict__ W1, const float* __restrict__ xn,
                       float* __restrict__ h1) {
  int wave = (blockIdx.x * 256 + threadIdx.x) >> 5;
  int lane = threadIdx.x & 31;
  int mT = wave & 15;          // 16 tiles of 32 rows over 512
  int colT = wave >> 4;        // 1024 column tiles of 32
  int m = lane & 15, kh = lane >> 4, nl = lane & 15;
  int m0 = mT * 32;
  int col0 = colT * 32;
  int n = col0 >> 12;
  int pix0 = (col0 & 4095) + nl;
  size_t bb0 = (size_t)n * 524288 + pix0;
  size_t bb1 = bb0 + 16;
  const float* ar0 = W1 + (size_t)(m0 + m) * 128;
  const float* ar1 = ar0 + 16 * 128;
  v8f a00 = {}, a01 = {}, a10 = {}, a11 = {};
  for (int k0 = 0; k0 < 128; k0 += 4) {
    int c = k0 + 2 * kh;
    __builtin_prefetch(&xn[bb0 + (size_t)(c + 16) * 4096], 0, 1);
    v2f A0; A0.x = ar0[c]; A0.y = ar0[c + 1];
    v2f A1; A1.x = ar1[c]; A1.y = ar1[c + 1];
    v2f B0; B0.x = xn[bb0 + (size_t)c * 4096];
    B0.y = xn[bb0 + (size_t)(c + 1) * 4096];
    v2f B1; B1.x = xn[bb1 + (size_t)c * 4096];
    B1.y = xn[bb1 + (size_t)(c + 1) * 4096];
    a00 = wmma4(A0, B0, a00);
    a01 = wmma4(A0, B1, a01);
    a10 = wmma4(A1, B0, a10);
    a11 = wmma4(A1, B1, a11);
  }
  size_t obase = (size_t)n * 2097152 + pix0;
#pragma unroll
  for (int r = 0; r < 8; ++r) {
    int oc0 = m0 + r + 8 * kh;
    int oc1 = oc0 + 16;
    float h;
    h = a00[r]; h1[obase + (size_t)oc0 * 4096]      = 0.5f * h * (1.f + erff(h * 0.70710678118f));
    h = a01[r]; h1[obase + (size_t)oc0 * 4096 + 16] = 0.5f * h * (1.f + erff(h * 0.70710678118f));
    h = a10[r]; h1[obase + (size_t)oc1 * 4096]      = 0.5f * h * (1.f + erff(h * 0.70710678118f));
    h = a11[r]; h1[obase + (size_t)oc1 * 4096 + 16] = 0.5f * h * (1.f + erff(h * 0.70710678118f));
  }
}

// ===========================================================================
// 17) out = mlp_w2 @ h1 + o   WMMA GEMM 2x2-blocked, M=128, K=512
// ===========================================================================
__global__ void k_mlp2(const float* __restrict__ W2, const float* __restrict__ h1,
                       const float* __restrict__ o, float* __restrict__ out) {
  int wave = (blockIdx.x * 256 + threadIdx.x) >> 5;
  int lane = threadIdx.x & 31;
  int mT = wave & 3;           // 4 tiles of 32 rows over 128
  int colT = wave >> 2;        // 1024 column tiles of 32
  int m = lane & 15, kh = lane >> 4, nl = lane & 15;
  int m0 = mT * 32;
  int col0 = colT * 32;
  int n = col0 >> 12;
  int pix0 = (col0 & 4095) + nl;
  size_t bb0 = (size_t)n * 2097152 + pix0;
  size_t bb1 = bb0 + 16;
  const float* ar0 = W2 + (size_t)(m0 + m) * 512;
  const float* ar1 = ar0 + 16 * 512;
  v8f a00 = {}, a01 = {}, a10 = {}, a11 = {};
  for (int k0 = 0; k0 < 512; k0 += 4) {
    int c = k0 + 2 * kh;
    __builtin_prefetch(&h1[bb0 + (size_t)(c + 16) * 4096], 0, 1);
    v2f A0; A0.x = ar0[c]; A0.y = ar0[c + 1];
    v2f A1; A1.x = ar1[c]; A1.y = ar1[c + 1];
    v2f B0; B0.x = h1[bb0 + (size_t)c * 4096];
    B0.y = h1[bb0 + (size_t)(c + 1) * 4096];
    v2f B1; B1.x = h1[bb1 + (size_t)c * 4096];
    B1.y = h1[bb1 + (size_t)(c + 1) * 4096];
    a00 = wmma4(A0, B0, a00);
    a01 = wmma4(A0, B1, a01);
    a10 = wmma4(A1, B0, a10);
    a11 = wmma4(A1, B1, a11);
  }
  size_t obase = (size_t)n * 524288 + pix0;
#pragma unroll
  for (int r = 0; r < 8; ++r) {
    int oc0 = m0 + r + 8 * kh;
    int oc1 = oc0 + 16;
    size_t i00 = obase + (size_t)oc0 * 4096;
    size_t i01 = i00 + 16;
    size_t i10 = obase + (size_t)oc1 * 4096;
    size_t i11 = i10 + 16;
    out[i00] = a00[r] + o[i00];
    out[i01] = a01[r] + o[i01];
    out[i10] = a10[r] + o[i10];
    out[i11] = a11[r] + o[i11];
  }
}

// ===========================================================================
// launch
// ===========================================================================
extern "C" void kernel_launch(void* const* d_in, const int* in_sizes, int n_in,
                              void* d_out, int out_size, void* d_ws,
                              size_t ws_size, hipStream_t stream) {
  const float* x      = (const float*)d_in[0];
  const float* qkv_w  = (const float*)d_in[1];
  const float* bqg    = (const float*)d_in[2];
  const float* bqb    = (const float*)d_in[3];
  const float* brel   = (const float*)d_in[4];
  const float* bsg    = (const float*)d_in[5];
  const float* bsb    = (const float*)d_in[6];
  const float* bog    = (const float*)d_in[7];
  const float* bob    = (const float*)d_in[8];
  const float* ing    = (const float*)d_in[9];
  const float* inb    = (const float*)d_in[10];
  const float* w1     = (const float*)d_in[11];
  const float* w2     = (const float*)d_in[12];
  float* out = (float*)d_out;

  float* ws = (float*)d_ws;
  float* qkv   = ws;                  // 8388608
  float* qstat = qkv   + 8388608;     // 512
  float* pos   = qstat + 512;         // 34848
  float* qa    = pos   + 34848;       // 1081344
  float* ka    = qa    + 1081344;     // 1081344
  float* va    = ka    + 1081344;     // 2162688
  float* simp  = va    + 2162688;     // 13381632
  float* sstat = simp  + 13381632;    // 384
  float* simG  = sstat + 384;         // 4460544
  float* simH  = simG  + 4460544;     // 16777216
  float* vb    = simH  + 16777216;    // 4194304
  float* veH   = vb    + 4194304;     // 65536
  float* so    = veH   + 65536;       // 8388608
  float* sost  = so    + 8388608;     // 512
  float* obuf  = sost  + 512;         // 4194304
  float* inst  = obuf  + 4194304;     // 2048
  float* xn    = inst  + 2048;        // 4194304
  float* h1    = xn    + 4194304;     // 16777216

  k_qkv_gemm<<<1024, 256, 0, stream>>>(qkv_w, x, qkv);
  k_chan_stats<<<256, 256, 0, stream>>>(qkv, qstat, 256, 512, 64);
  k_pos<<<137, 256, 0, stream>>>(brel, pos);
  k_split_resize<<<16896, 256, 0, stream>>>(qkv, qstat, bqg, bqb, qa, ka, va);
  k_sim_pre<<<52272, 256, 0, stream>>>(qa, ka, pos, simp);
  k_sim_stats<<<192, 256, 0, stream>>>(simp, sstat);
  k_simG<<<17424, 256, 0, stream>>>(simp, sstat, bsg, bsb, simG);
  k_resize_softmax<<<1024, 256, 0, stream>>>(simG, simH);
  k_vb<<<16384, 256, 0, stream>>>(va, vb);
  k_veH<<<256, 256, 0, stream>>>(pos, veH);
  k_sv<<<1024, 256, 0, stream>>>(simH, vb, so);
  k_sve<<<16384, 256, 0, stream>>>(simH, veH, so);
  k_chan_stats<<<256, 256, 0, stream>>>(so, sost, 256, 512, 64);
  k_o<<<16384, 256, 0, stream>>>(so, sost, bog, bob, obuf);
  k_in_stats<<<1024, 256, 0, stream>>>(obuf, inst);
  k_xn<<<16384, 256, 0, stream>>>(obuf, inst, ing, inb, xn);
  k_mlp1<<<2048, 256, 0, stream>>>(w1, xn, h1);
  k_mlp2<<<512, 256, 0, stream>>>(w2, h1, obuf, out);
}